// Feather_statistic_append_35442070126678
// MI455X (gfx1250) — compile-verified
//
#include <hip/hip_runtime.h>
#include <math.h>

typedef __attribute__((ext_vector_type(2))) float v2f;
typedef __attribute__((ext_vector_type(8))) float v8f;

#define B_ROWS 256
#define D_DIM  2048
#define T_K    10.0f

// ---------------------------------------------------------------------------
// Kernel 1: per-row mean / unbiased std, rowterm = mean^2 + std^2,
//           and (re)initialize per-row min-distance^2 to +inf.
// ---------------------------------------------------------------------------
__global__ __launch_bounds__(256) void row_stats_kernel(
    const float* __restrict__ feat,
    float* __restrict__ means,
    float* __restrict__ stds,
    float* __restrict__ rowterm,
    unsigned int* __restrict__ minsq)
{
    __shared__ float s_sum[256];
    __shared__ float s_sq[256];

    const int b   = blockIdx.x;
    const int tid = threadIdx.x;
    const float* __restrict__ row = feat + (size_t)b * D_DIM;

    float sum = 0.f, sq = 0.f;
#pragma unroll
    for (int i = 0; i < D_DIM / 256; ++i) {
        float x = row[tid + i * 256];
        sum += x;
        sq   = fmaf(x, x, sq);
    }
    s_sum[tid] = sum;
    s_sq[tid]  = sq;
    __syncthreads();

    for (int off = 128; off > 0; off >>= 1) {
        if (tid < off) {
            s_sum[tid] += s_sum[tid + off];
            s_sq[tid]  += s_sq[tid + off];
        }
        __syncthreads();
    }

    if (tid == 0) {
        float mean = s_sum[0] * (1.0f / (float)D_DIM);
        float var  = (s_sq[0] - (float)D_DIM * mean * mean) * (1.0f / (float)(D_DIM - 1));
        var        = fmaxf(var, 0.0f);
        float sd   = sqrtf(var);
        means[b]   = mean;
        stds[b]    = sd;
        rowterm[b] = mean * mean + sd * sd;
        minsq[b]   = 0x7F800000u;   // +inf bits
    }
}

// ---------------------------------------------------------------------------
// Kernel 2: pairwise min distance^2 via rank-2 WMMA.
//   A = [-2*mean | -2*std] (16x4, K=2,3 zeroed),  B = [mu ; sigma] (4x16),
//   C = rowterm broadcast  ->  D = rowterm[M] - 2*(m*mu + s*sig)
//   dist^2 = D + (mu^2 + sig^2)
// Running min kept as uint bit-pattern (dist^2 >= 0 after clamp), so the
// loop-carried op is a bare v_min_u32 (no float canonicalization).
// ---------------------------------------------------------------------------
__global__ __launch_bounds__(256) void pair_min_kernel(
    const float* __restrict__ qmu,
    const float* __restrict__ qsig,
    const float* __restrict__ means,
    const float* __restrict__ stds,
    const float* __restrict__ rowterm,
    unsigned int* __restrict__ minsq,
    int Q)
{
    __shared__ float s_m[B_ROWS];
    __shared__ float s_s[B_ROWS];
    __shared__ float s_rt[B_ROWS];

    const int tid = threadIdx.x;
    s_m[tid]  = means[tid];
    s_s[tid]  = stds[tid];
    s_rt[tid] = rowterm[tid];
    __syncthreads();

    const int wave  = tid >> 5;            // 0..7
    const int lane  = tid & 31;
    const int half  = lane >> 4;           // 0: lanes 0-15, 1: lanes 16-31
    const int ln16  = lane & 15;
    const int btile = blockIdx.y * 8 + wave;   // 0..15 (gridDim.y == 2)

    // A fragment (loop invariant): -2*[mean|std] rows; K=2,3 half zeroed.
    const float m = s_m[(btile << 4) + ln16];
    const float s = s_s[(btile << 4) + ln16];
    v2f a;
    a.x = half ? 0.0f : (-2.0f * m);
    a.y = half ? 0.0f : (-2.0f * s);

    // C input = rowterm per accumulator VGPR (loop invariant).
    const int mbase = (btile << 4) + (half << 3);
    v8f crt;
#pragma unroll
    for (int r = 0; r < 8; ++r) crt[r] = s_rt[mbase + r];

    unsigned int mnu[8];
#pragma unroll
    for (int r = 0; r < 8; ++r) mnu[r] = 0x7F800000u;   // +inf bits

    // ---- main loop: full 16-wide q tiles, no bounds checks ----
    const int numFull = Q >> 4;
#pragma unroll 2
    for (int qt = blockIdx.x; qt < numFull; qt += gridDim.x) {
        const int q  = (qt << 4) + ln16;
        const float mu = qmu[q];
        const float sg = qsig[q];
        const float ct = fmaf(mu, mu, sg * sg);

        // No masking of B: A's K=2,3 entries are zero, so those products vanish.
        v2f bf;
        bf.x = mu;
        bf.y = sg;

        v8f c = __builtin_amdgcn_wmma_f32_16x16x4_f32(
                    false, a, false, bf, (short)0, crt, false, false);

#pragma unroll
        for (int r = 0; r < 8; ++r) {
            const float d = fmaxf(ct + c[r], 0.0f);      // dist^2, clamped >= 0
            const unsigned int du = __float_as_uint(d);
            mnu[r] = (du < mnu[r]) ? du : mnu[r];        // v_min_u32
        }
    }

    // ---- tail: one partial tile, handled by a single designated block ----
    if ((Q & 15) && blockIdx.x == (unsigned)(numFull % (int)gridDim.x)) {
        const int q  = (numFull << 4) + ln16;
        const int qc = q < Q ? q : (Q - 1);              // clamp keeps EXEC uniform
        const float mu = qmu[qc];
        const float sg = qsig[qc];
        const float ct = (q < Q) ? fmaf(mu, mu, sg * sg) : __builtin_inff();

        v2f bf;
        bf.x = mu;
        bf.y = sg;

        v8f c = __builtin_amdgcn_wmma_f32_16x16x4_f32(
                    false, a, false, bf, (short)0, crt, false, false);

#pragma unroll
        for (int r = 0; r < 8; ++r) {
            const float d = fmaxf(ct + c[r], 0.0f);
            const unsigned int du = __float_as_uint(d);
            mnu[r] = (du < mnu[r]) ? du : mnu[r];
        }
    }

    // Min across the 16 lanes of each half (halves stay separate under xor<16).
#pragma unroll
    for (int r = 0; r < 8; ++r) {
        unsigned int v = mnu[r];
        unsigned int t;
        t = (unsigned int)__shfl_xor((int)v, 1, 32); v = (t < v) ? t : v;
        t = (unsigned int)__shfl_xor((int)v, 2, 32); v = (t < v) ? t : v;
        t = (unsigned int)__shfl_xor((int)v, 4, 32); v = (t < v) ? t : v;
        t = (unsigned int)__shfl_xor((int)v, 8, 32); v = (t < v) ? t : v;
        if (ln16 == 0) {
            atomicMin(&minsq[mbase + r], v);
        }
    }
}

// ---------------------------------------------------------------------------
// Kernel 3: T = exp(-T_k * sqrt(min_dist^2))
// ---------------------------------------------------------------------------
__global__ __launch_bounds__(256) void finalize_kernel(
    const unsigned int* __restrict__ minsq, float* __restrict__ out)
{
    const int i = threadIdx.x;
    const float d2 = __uint_as_float(minsq[i]);
    out[i] = expf(-T_K * sqrtf(d2));
}

// ---------------------------------------------------------------------------
extern "C" void kernel_launch(void* const* d_in, const int* in_sizes, int n_in,
                              void* d_out, int out_size, void* d_ws, size_t ws_size,
                              hipStream_t stream)
{
    const float* feat = (const float*)d_in[0];   // features [256, 2048] f32
    const float* qmu  = (const float*)d_in[4];   // queue_mus [Q] f32
    const float* qsig = (const float*)d_in[5];   // queue_sigmas [Q] f32
    const int Q = in_sizes[4];
    float* out = (float*)d_out;

    // Workspace layout: means | stds | rowterm | minsq  (4 KB total)
    float*        ws_m   = (float*)d_ws;
    float*        ws_s   = ws_m + B_ROWS;
    float*        ws_rt  = ws_s + B_ROWS;
    unsigned int* ws_min = (unsigned int*)(ws_rt + B_ROWS);

    row_stats_kernel<<<B_ROWS, 256, 0, stream>>>(feat, ws_m, ws_s, ws_rt, ws_min);

    dim3 grid(1024, 2, 1);   // x: q-chunks, y: 2 * 8 waves = 16 b-tiles
    pair_min_kernel<<<grid, 256, 0, stream>>>(qmu, qsig, ws_m, ws_s, ws_rt, ws_min, Q);

    finalize_kernel<<<1, B_ROWS, 0, stream>>>(ws_min, out);
}